// Slater_15942918602892
// MI455X (gfx1250) — compile-verified
//
#include <hip/hip_runtime.h>
#include <math.h>

// Slater log-determinant evaluation for MI455X (gfx1250, wave32).
//   B=1024 samples, M=16 determinants, Ne=64 electrons/spin, Nsites=128.
// Kernel 1: ONE wave32 per 64x64 matrix, single-wave workgroups so all
//   barriers are hardware NOPs (ISA 3.1). Row-gather into LDS, blocked LU
//   (NB=16) with partial pivoting; Schur update A22 -= L21*U12 runs on the
//   matrix pipe via V_WMMA_F32_16X16X4_F32. L21 is stored pre-negated so
//   the WMMA A-fragment needs no per-op negation (FP32 WMMA lacks A-neg).
// Kernel 2: complex logsumexp over M per sample (signs +-1 => signed real sum).

typedef __attribute__((ext_vector_type(2))) float v2f;
typedef __attribute__((ext_vector_type(8))) float v8f;

#define NSITES 128
#define NE     64
#define MDET   16
#define BSAMP  1024
// Bank model: word address i*STRIDE+j -> bank (i+j)%64 with STRIDE=65:
// conflict-free for column walks (consecutive rows) AND row walks
// (consecutive cols) -- covers pivot search, swaps, rank-1, U12, WMMA frags.
#define STRIDE 65

__global__ __launch_bounds__(32) void slater_lu_kernel(
    const float* __restrict__ U_up, const float* __restrict__ U_down,
    const int* __restrict__ y_up, const int* __restrict__ y_down,
    float2* __restrict__ partial)
{
    __shared__ float sA[NE * STRIDE];   // one padded 64x64 matrix
    __shared__ int   sIdx[NE];

    const int lane = threadIdx.x;
    const int bms  = blockIdx.x;        // (b*M + m)*2 + spin
    const int spin = bms & 1;
    const int bm   = bms >> 1;
    const int b    = bm / MDET;
    const int m    = bm % MDET;

    const float* __restrict__ U = spin ? U_down : U_up;
    const int*   __restrict__ y = spin ? y_down : y_up;
    float* A = sA;

    // ---- load electron-site indices for this sample ----
    sIdx[lane]      = y[b * NE + lane];
    sIdx[lane + 32] = y[b * NE + lane + 32];
    __syncthreads();                    // single-wave: barrier is a NOP + fence

    // ---- gather 64 rows of U[m]: float4 global loads, b32 LDS stores ----
    for (int t = lane; t < NE * 16; t += 32) {
        const int r = t >> 4;           // matrix row
        const int q = t & 15;           // float4 chunk within row
        const float4* src = reinterpret_cast<const float4*>(
            U + ((size_t)m * NSITES + sIdx[r]) * NE);
        float4 v = src[q];
        float* dst = &A[r * STRIDE + q * 4];
        dst[0] = v.x; dst[1] = v.y; dst[2] = v.z; dst[3] = v.w;
    }
    __syncthreads();

    float logabs = 0.0f;
    float sgn    = 1.0f;

    const int lane_lo = lane & 15;
    const int lane_hi = lane >> 4;

    for (int kb = 0; kb < 4; ++kb) {
        const int c0   = kb * 16;
        const int cend = c0 + 16;

        // ===== panel factorization (columns c0..c0+15), partial pivoting =====
        for (int c = c0; c < cend; ++c) {
            __syncthreads();
            // pivot search: argmax |A[i][c]|, i in [c, 64)
            float best = -1.0f;
            int   brow = c;
            for (int i = c + lane; i < NE; i += 32) {
                float v = fabsf(A[i * STRIDE + c]);
                if (v > best) { best = v; brow = i; }
            }
            for (int off = 16; off > 0; off >>= 1) {
                float ob   = __shfl_xor(best, off);
                int   orow = __shfl_xor(brow, off);
                if (ob > best || (ob == best && orow < brow)) { best = ob; brow = orow; }
            }
            const int p = brow;
            __syncthreads();
            if (p != c) {                       // full-width row swap
                for (int col = lane; col < NE; col += 32) {
                    float tmp           = A[p * STRIDE + col];
                    A[p * STRIDE + col] = A[c * STRIDE + col];
                    A[c * STRIDE + col] = tmp;
                }
            }
            __syncthreads();
            const float piv = A[c * STRIDE + c];
            if (p != c)      sgn = -sgn;
            if (piv < 0.0f)  sgn = -sgn;
            logabs += logf(fabsf(piv));
            const float inv = 1.0f / piv;
            // store NEGATED multipliers; panel rank-1 becomes an FMA-add
            for (int i = c + 1 + lane; i < NE; i += 32) {
                float nm = -(A[i * STRIDE + c] * inv);
                A[i * STRIDE + c] = nm;         // holds -L21
                for (int j = c + 1; j < cend; ++j)
                    A[i * STRIDE + j] += nm * A[c * STRIDE + j];
            }
        }
        __syncthreads();

        const int t0 = cend;                    // uniform trip structure
        if (t0 < NE) {
            // ===== U12 = L11^-1 * A12 : lanes own trailing columns =====
            // (stored panel entries are -L, so forward-subst adds)
            for (int col = t0 + lane; col < NE; col += 32) {
                for (int r = c0 + 1; r < cend; ++r) {
                    float s = A[r * STRIDE + col];
                    for (int t = c0; t < r; ++t)
                        s += A[r * STRIDE + t] * A[t * STRIDE + col];
                    A[r * STRIDE + col] = s;
                }
            }
            __syncthreads();

            // ===== A22 += (-L21) * U12 via WMMA F32 16x16x4 =====
            const int nt = (NE - t0) >> 4;
            for (int ti = 0; ti < nt; ++ti) {
                for (int tj = 0; tj < nt; ++tj) {
                    const int r0   = t0 + ti * 16;
                    const int col0 = t0 + tj * 16;
                    v8f cfrag;
                    #pragma unroll
                    for (int v = 0; v < 8; ++v)   // C layout: M = v + 8*lane[4]
                        cfrag[v] = A[(r0 + v + lane_hi * 8) * STRIDE + col0 + lane_lo];
                    #pragma unroll
                    for (int kk = 0; kk < 4; ++kk) {
                        const int k0 = c0 + kk * 4;
                        v2f afrag, bfrag;
                        // A 16x4: row = lane[3:0]; K pair = 2*lane[4] (holds -L21)
                        const int ar = r0 + lane_lo;
                        const int ak = k0 + lane_hi * 2;
                        afrag[0] = A[ar * STRIDE + ak];
                        afrag[1] = A[ar * STRIDE + ak + 1];
                        // B 4x16: col = lane[3:0]; K pair = 2*lane[4]
                        const int bc = col0 + lane_lo;
                        const int bk = k0 + lane_hi * 2;
                        bfrag[0] = A[bk * STRIDE + bc];
                        bfrag[1] = A[(bk + 1) * STRIDE + bc];
                        cfrag = __builtin_amdgcn_wmma_f32_16x16x4_f32(
                            false, afrag, false, bfrag, (short)0, cfrag,
                            false, false);
                    }
                    #pragma unroll
                    for (int v = 0; v < 8; ++v)
                        A[(r0 + v + lane_hi * 8) * STRIDE + col0 + lane_lo] = cfrag[v];
                }
            }
            __syncthreads();
        }
    }

    if (lane == 0)
        partial[bms] = make_float2(logabs, sgn);   // per (b, m, spin)
}

// Complex logsumexp over M determinants per sample. Signs are +-1, so
// imag parts are {0, pi} and the exp-sum collapses to a signed real sum.
__global__ __launch_bounds__(32) void slater_lse_kernel(
    const float2* __restrict__ partial, float2* __restrict__ out)
{
    const int b    = blockIdx.x;
    const int lane = threadIdx.x;
    float x = -INFINITY, s = 1.0f;
    if (lane < MDET) {
        float2 pu = partial[(b * MDET + lane) * 2 + 0];
        float2 pd = partial[(b * MDET + lane) * 2 + 1];
        x = pu.x + pd.x;        // log|det_up| + log|det_dn|
        s = pu.y * pd.y;        // combined sign
    }
    float mx = x;
    for (int off = 16; off > 0; off >>= 1)
        mx = fmaxf(mx, __shfl_xor(mx, off));
    float v = (lane < MDET) ? s * expf(x - mx) : 0.0f;
    for (int off = 16; off > 0; off >>= 1)
        v += __shfl_xor(v, off);
    if (lane == 0) {
        float re, im;
        if (v > 0.0f)      { re = mx + logf(v);  im = 0.0f; }
        else if (v < 0.0f) { re = mx + logf(-v); im = 3.14159265358979323846f; }
        else               { re = -INFINITY;     im = 0.0f; }
        out[b] = make_float2(re, im);
    }
}

extern "C" void kernel_launch(void* const* d_in, const int* in_sizes, int n_in,
                              void* d_out, int out_size, void* d_ws, size_t ws_size,
                              hipStream_t stream) {
    const float* U_up   = (const float*)d_in[0];
    const float* U_down = (const float*)d_in[1];
    const int*   y_up   = (const int*)d_in[2];
    const int*   y_down = (const int*)d_in[3];
    float2* partial = (float2*)d_ws;          // B*M*2 float2 = 256 KB scratch

    slater_lu_kernel<<<BSAMP * MDET * 2, 32, 0, stream>>>(
        U_up, U_down, y_up, y_down, partial);
    slater_lse_kernel<<<BSAMP, 32, 0, stream>>>(
        partial, (float2*)d_out);
}